// RelaxedConvolution_58815282151654
// MI455X (gfx1250) — compile-verified
//
#include <hip/hip_runtime.h>

// ---------------------------------------------------------------------------
// RelaxedConvolution (e3nn-style scalar tensor product) fused kernel, gfx1250
//
//   h   = sqrt(2) * relu(S @ W1/4)          [E,64]
//   tpw = h @ W2/8                          [E,1024]   (never materialized)
//   f[e,k] = (1/8) * sum_{ij} x[e,i]*a[e,j] * tpw[e, ij*16+k]
//   out[n,k] = (1/4) * segment_sum(f, edge_dst)
//
// Dominant math: E x ([1x64]@[64x1024]) = 42 GFLOP fp32 -> v_wmma_f32_16x16x4_f32.
// W2 (256 KB, raw) is DMA'd into LDS by the Tensor Data Mover (tensor_load_to_lds,
// TENSORcnt-tracked); its 1/sqrt(64) normalization is linear and folded into the
// final scatter scale: (1/8 fc2) * (1/8 path) * (1/4 neighbors) = 1/256.
// ---------------------------------------------------------------------------

typedef __attribute__((ext_vector_type(2))) float v2f;
typedef __attribute__((ext_vector_type(8))) float v8f;
typedef __attribute__((ext_vector_type(4))) unsigned int v4u;
typedef __attribute__((ext_vector_type(8))) unsigned int v8u;

#define N_NODES_C 20000
#define N_EDGES_C 320000
#define MUL_IN_C 16
#define MUL_EDGE_C 4
#define MUL_OUT_C 16
#define FC_DIM_C 16
#define FC_HID_C 64
#define WNUMEL_C 1024
#define TILES_C (N_EDGES_C / 16)

// LDS layout in floats:
//   [0, 1024)                : W1 (16x64), pre-scaled by 1/sqrt(16)
//   [1024, 1024+65536)       : W2 (64x1024), RAW (TDM copy; scale folded out)
//   [66560 + wave*1024, ...) : per-wave 16x64 scratch (H tile, then o tile)
#define LDS_W1_OFF 0
#define LDS_W2_OFF 1024
#define LDS_WV_OFF (1024 + 65536)
#define LDS_FLOATS (1024 + 65536 + 8 * 1024)

__global__ void zero_out_kernel(float* __restrict__ out, int n) {
    int i = blockIdx.x * blockDim.x + threadIdx.x;
    if (i < n) out[i] = 0.0f;
}

__global__ void __launch_bounds__(256, 1)
relaxed_conv_fused_kernel(const float* __restrict__ node_input,
                          const float* __restrict__ edge_attr,
                          const float* __restrict__ edge_scalars,
                          const float* __restrict__ fc_w1,
                          const float* __restrict__ fc_w2,
                          const int*   __restrict__ edge_src,
                          const int*   __restrict__ edge_dst,
                          float*       __restrict__ out)
{
    extern __shared__ float lds[];
    float* w1l = lds + LDS_W1_OFF;
    float* w2l = lds + LDS_W2_OFF;

    const int tid  = threadIdx.x;
    const int lane = tid & 31;
    const int wave = tid >> 5;
    const int half = lane >> 4;   // which 16-lane half
    const int ln   = lane & 15;

    float* wbuf = lds + LDS_WV_OFF + wave * 1024;  // per-wave 16x64 scratch

    // ---- W1 staged by plain loads (tiny), scaled by 1/sqrt(16)=0.25 ----
    for (int i = tid; i < FC_DIM_C * FC_HID_C; i += 256)
        w1l[i] = fc_w1[i] * 0.25f;

    // ---- W2 staged RAW by the Tensor Data Mover: 2-D tensor 64 x 1024 f32 ----
    // D# group0: count=1 | lds_addr | global_addr | type=2
    // D# group1: data_size=4B, tensor_dim0=1024, tensor_dim1=64,
    //            tile_dim0=1024, tile_dim1=64, tensor_dim0_stride=1024
    if (wave == 0) {
        unsigned long long ga = (unsigned long long)(const void*)fc_w2;
        v4u g0;
        g0[0] = 1u;                                   // count=1 (valid descriptor)
        g0[1] = (unsigned)(LDS_W2_OFF * 4);           // lds_addr (bytes)
        g0[2] = (unsigned)(ga & 0xFFFFFFFFu);         // global_addr[31:0]
        g0[3] = (unsigned)((ga >> 32) & 0x1FFFFFFu)   // global_addr[56:32]
              | (2u << 30);                           // type=2 ("image")
        v8u g1;
        g1[0] = (2u << 16);                           // data_size=2 -> 4 bytes
        g1[1] = (unsigned)(WNUMEL_C) << 16;           // tensor_dim0[15:0] @ bits 63:48
        g1[2] = (unsigned)(FC_HID_C) << 16;           // tensor_dim1[15:0] @ bits 95:80
        g1[3] = (unsigned)(WNUMEL_C) << 16;           // tile_dim0 @ bits 127:112
        g1[4] = (unsigned)(FC_HID_C);                 // tile_dim1 @ bits 143:128
        g1[5] = (unsigned)(WNUMEL_C);                 // tensor_dim0_stride[31:0]
        g1[6] = 0u;
        g1[7] = 0u;
        v4u g2 = (v4u)(0u);                           // dims 2/3 unused (tile_dim=0)
        v4u g3 = (v4u)(0u);
        asm volatile("tensor_load_to_lds %0, %1, %2, %3"
                     :: "s"(g0), "s"(g1), "s"(g2), "s"(g3)
                     : "memory");
        __builtin_amdgcn_s_wait_tensorcnt(0);
    }
    __syncthreads();

    const int waves_total = gridDim.x * 8;
    const int gwave       = blockIdx.x * 8 + wave;

    for (int tile = gwave; tile < TILES_C; tile += waves_total) {
        const int e0 = tile * 16;

        // prefetch next tile's FC inputs into the cache hierarchy
        {
            const int nt = tile + waves_total;
            if (nt < TILES_C)
                __builtin_prefetch(edge_scalars + (size_t)nt * 16 * FC_DIM_C, 0, 1);
        }

        // ================= FC layer: H = sqrt2*relu(S @ W1s) =================
        // A-fragments of S: M=edge(16), K=fc_dim(16) -> 4 K-steps.
        // A layout (ISA 7.12.2): lanes 0-15 M=0..15 hold K={0,1}; lanes 16-31 K={2,3}.
        v2f sa[4];
        {
            const float* sp = edge_scalars + (size_t)(e0 + ln) * FC_DIM_C;
            #pragma unroll
            for (int kk = 0; kk < 4; ++kk) {
                const int kb = kk * 4 + half * 2;
                sa[kk].x = sp[kb];
                sa[kk].y = sp[kb + 1];
            }
        }
        #pragma unroll
        for (int nt = 0; nt < 4; ++nt) {
            v8f acc = {};
            #pragma unroll
            for (int kk = 0; kk < 4; ++kk) {
                const int row0 = kk * 4 + half * 2;
                v2f b;
                b.x = w1l[row0 * FC_HID_C + nt * 16 + ln];
                b.y = w1l[(row0 + 1) * FC_HID_C + nt * 16 + ln];
#if __has_builtin(__builtin_amdgcn_wmma_f32_16x16x4_f32)
                acc = __builtin_amdgcn_wmma_f32_16x16x4_f32(
                    false, sa[kk], false, b, (short)0, acc, false, false);
#else
                acc[0] += sa[kk].x * b.x + sa[kk].y * b.y;  // placeholder fallback
#endif
            }
            // relu * sqrt2; spill H tile to LDS as [M][64] (D layout -> row major)
            #pragma unroll
            for (int r = 0; r < 8; ++r) {
                const int M = r + half * 8;
                float h = fmaxf(acc[r], 0.0f) * 1.41421356237f;
                wbuf[M * FC_HID_C + nt * 16 + ln] = h;
            }
        }

        // A-fragments of H for the big GEMM: M=edge, K=h(64) -> 16 K-steps.
        // (wave-internal DS ops are in-order; no barrier needed)
        v2f ha[16];
        #pragma unroll
        for (int kk = 0; kk < 16; ++kk) {
            const int kb = kk * 4 + half * 2;
            ha[kk].x = wbuf[ln * FC_HID_C + kb];
            ha[kk].y = wbuf[ln * FC_HID_C + kb + 1];
        }

        // ============== outer product o[e,ij] = x[e,i]*a[e,j] -> wbuf ==============
        {
            const int e   = e0 + ln;
            const int s   = edge_src[e];
            const float* xp = node_input + (size_t)s * MUL_IN_C;
            const float* ap = edge_attr + (size_t)e * MUL_EDGE_C;
            const float a0 = ap[0], a1 = ap[1], a2 = ap[2], a3 = ap[3];
            #pragma unroll
            for (int ii = 0; ii < 8; ++ii) {
                const int i  = half * 8 + ii;
                const float xv = xp[i];
                wbuf[ln * 64 + i * 4 + 0] = xv * a0;
                wbuf[ln * 64 + i * 4 + 1] = xv * a1;
                wbuf[ln * 64 + i * 4 + 2] = xv * a2;
                wbuf[ln * 64 + i * 4 + 3] = xv * a3;
            }
        }

        // ========== main GEMM (tpw tile) + fold with o, tpw never stored ==========
        v8f F = {};
        #pragma unroll 4
        for (int t = 0; t < 64; ++t) {          // N-tile t == flat ij index
            v8f acc = {};
            #pragma unroll
            for (int kk = 0; kk < 16; ++kk) {
                const int row0 = kk * 4 + half * 2;
                v2f b;
                b.x = w2l[row0 * WNUMEL_C + t * 16 + ln];
                b.y = w2l[(row0 + 1) * WNUMEL_C + t * 16 + ln];
#if __has_builtin(__builtin_amdgcn_wmma_f32_16x16x4_f32)
                acc = __builtin_amdgcn_wmma_f32_16x16x4_f32(
                    false, ha[kk], false, b, (short)0, acc, false, false);
#else
                acc[0] += ha[kk].x * b.x + ha[kk].y * b.y;
#endif
            }
            #pragma unroll
            for (int r = 0; r < 8; ++r) {
                const int M = r + half * 8;
                F[r] += acc[r] * wbuf[M * 64 + t];   // broadcast LDS read per half
            }
        }

        // == scatter-add, scale = (1/8 fc2) * (1/8 path) * (1/4 neighbors) = 1/256 ==
        #pragma unroll
        for (int r = 0; r < 8; ++r) {
            const int M = r + half * 8;
            const int d = edge_dst[e0 + M];
            atomicAdd(&out[(size_t)d * MUL_OUT_C + ln], F[r] * 0.00390625f);
        }
    }
}

extern "C" void kernel_launch(void* const* d_in, const int* in_sizes, int n_in,
                              void* d_out, int out_size, void* d_ws, size_t ws_size,
                              hipStream_t stream) {
    (void)in_sizes; (void)n_in; (void)d_ws; (void)ws_size;

    const float* node_input   = (const float*)d_in[0];
    const float* edge_attr    = (const float*)d_in[1];
    const float* edge_scalars = (const float*)d_in[2];
    const float* fc_w1        = (const float*)d_in[3];
    const float* fc_w2        = (const float*)d_in[4];
    const int*   edge_src     = (const int*)d_in[5];
    const int*   edge_dst     = (const int*)d_in[6];
    float*       out          = (float*)d_out;

    // output is accumulated with atomics -> zero it every call
    zero_out_kernel<<<(out_size + 255) / 256, 256, 0, stream>>>(out, out_size);

    const size_t shmem = (size_t)LDS_FLOATS * sizeof(float);  // 292 KB < 320 KB WGP LDS
    (void)hipFuncSetAttribute((const void*)relaxed_conv_fused_kernel,
                              hipFuncAttributeMaxDynamicSharedMemorySize,
                              (int)shmem);

    relaxed_conv_fused_kernel<<<640, 256, shmem, stream>>>(
        node_input, edge_attr, edge_scalars, fc_w1, fc_w2,
        edge_src, edge_dst, out);
}